// SparsePreprocessor_70557722738955
// MI455X (gfx1250) — compile-verified
//
#include <hip/hip_runtime.h>
#include <stdint.h>

#define PIVOTS    8192      // LDS pivot count (32 KB)
#define BLOCK     1024      // 32 waves per workgroup
#define GRID      1024
#define SEG_STEP0 128       // next_pow2(ceil(V/PIVOTS)); valid while S <= 128

typedef int v4i __attribute__((ext_vector_type(4)));

// Scalar fallback for tail elements: full pivot + segment search, returns lower_bound pos
__device__ __forceinline__ int remap_one_scalar(int x, const int* piv,
                                                const int* __restrict__ sorted_ids,
                                                int V, int S) {
    int lo = 0, hi = PIVOTS;
    while (lo < hi) { int m = (lo + hi) >> 1; if (piv[m] <= x) lo = m + 1; else hi = m; }
    int j = lo - 1;
    int base = (j < 0) ? 0 : j * S;
    int h = base + S; if (h > V) h = V;
    int l = base; if (l > V - 1) l = V - 1;
    while (l < h) { int m = (l + h) >> 1; if (sorted_ids[m] < x) l = m + 1; else h = m; }
    return l > V - 1 ? V - 1 : l;
}

extern "C" __global__ __launch_bounds__(BLOCK)
void sparse_remap2_kernel(const int* __restrict__ raw_values,
                          const int* __restrict__ raw_keys,
                          const int* __restrict__ sorted_ids,
                          const int* __restrict__ mapped_idx,
                          int* __restrict__ out_values,
                          int* __restrict__ out_keys,
                          int N, int V, int S)
{
    __shared__ int pivots[PIVOTS];

    // ---- Fill pivot table via CDNA5 async global->LDS loads (ASYNCcnt path) ----
    for (int i = threadIdx.x; i < PIVOTS; i += BLOCK) {
        int p = i * S;
        if (p > V - 1) p = V - 1;
        unsigned lds_off = (unsigned)(uintptr_t)(&pivots[i]);
        unsigned long long gaddr = (unsigned long long)(uintptr_t)(sorted_ids + p);
        asm volatile("global_load_async_to_lds_b32 %0, %1, off"
                     :: "v"(lds_off), "v"(gaddr) : "memory");
    }
    asm volatile("s_wait_asynccnt 0x0" ::: "memory");
    __syncthreads();

    const int tid    = blockIdx.x * BLOCK + threadIdx.x;
    const int stride = gridDim.x * BLOCK;
    const int n4     = N >> 2;

    const v4i* __restrict__ rv4 = (const v4i*)raw_values;
    const v4i* __restrict__ rk4 = (const v4i*)raw_keys;

    for (int g = tid; g < n4; g += stride) {
        if (g + stride < n4) {
            __builtin_prefetch(rv4 + g + stride, 0, 0);
            __builtin_prefetch(rk4 + g + stride, 0, 0);
        }
        // Streaming inputs: non-temporal (don't evict L2-resident search tables)
        const v4i v = __builtin_nontemporal_load(rv4 + g);
        const v4i k = __builtin_nontemporal_load(rk4 + g);
        int x[8] = { v.x, v.y, v.z, v.w, k.x, k.y, k.z, k.w };

        // ---- Stage 1: branchless LDS pivot search, 8 chains in lockstep ----
        // cnt[c] = number of pivots <= x[c]
        int cnt[8];
        #pragma unroll
        for (int c = 0; c < 8; ++c)
            cnt[c] = (pivots[(PIVOTS >> 1) - 1] <= x[c]) ? (PIVOTS >> 1) : 0;
        #pragma unroll
        for (int step = PIVOTS >> 2; step >= 1; step >>= 1) {
            #pragma unroll
            for (int c = 0; c < 8; ++c)
                cnt[c] += (pivots[cnt[c] + step - 1] <= x[c]) ? step : 0;
        }

        // ---- Segment bounds ----
        int lo[8], n[8];
        #pragma unroll
        for (int c = 0; c < 8; ++c) {
            int j = cnt[c] - 1;
            int base = (j < 0) ? 0 : j * S;
            int h = base + S; if (h > V) h = V;
            int l = base;     if (l > V - 1) l = V - 1;
            lo[c] = l;
            n[c]  = h - l;    // >= 1
        }

        // ---- Stage 2: branchless jump search in global (L2-resident),
        //      fixed 8 levels, 8 independent loads in flight per level ----
        int ofs[8];
        #pragma unroll
        for (int c = 0; c < 8; ++c) ofs[c] = 0;
        #pragma unroll
        for (int step = SEG_STEP0; step >= 1; step >>= 1) {
            #pragma unroll
            for (int c = 0; c < 8; ++c) {
                int nidx    = ofs[c] + step;
                int clamped = (nidx <= n[c]) ? nidx : n[c];
                int a       = sorted_ids[lo[c] + clamped - 1];
                ofs[c] += ((nidx <= n[c]) && (a < x[c])) ? step : 0;
            }
        }

        // ---- Gather (L2-resident, keep RT policy) + non-temporal streaming stores ----
        int r[8];
        #pragma unroll
        for (int c = 0; c < 8; ++c) {
            int p = lo[c] + ofs[c];
            if (p > V - 1) p = V - 1;
            r[c] = mapped_idx[p];
        }
        const int base = g << 2;
        #pragma unroll
        for (int c = 0; c < 4; ++c)
            __builtin_nontemporal_store(r[c], &out_values[base + c]);
        #pragma unroll
        for (int c = 0; c < 4; ++c)
            __builtin_nontemporal_store(r[4 + c], &out_keys[base + c]);
    }

    // ---- Tail (N % 4 != 0): scalar path ----
    for (int i = (n4 << 2) + tid; i < N; i += stride) {
        out_values[i] = mapped_idx[remap_one_scalar(raw_values[i], pivots, sorted_ids, V, S)];
        out_keys[i]   = mapped_idx[remap_one_scalar(raw_keys[i],   pivots, sorted_ids, V, S)];
    }
}

extern "C" void kernel_launch(void* const* d_in, const int* in_sizes, int n_in,
                              void* d_out, int out_size, void* d_ws, size_t ws_size,
                              hipStream_t stream) {
    // setup_inputs() order:
    //   0: raw_values (N int32)   1: raw_keys (N int32)   2: scores (N f32)
    //   3: offsets (B+1 int32)    4: sorted_ids (V int32) 5: mapped_idx (V int32)
    const int* raw_values = (const int*)d_in[0];
    const int* raw_keys   = (const int*)d_in[1];
    const float* scores   = (const float*)d_in[2];
    const int* offsets    = (const int*)d_in[3];
    const int* sorted_ids = (const int*)d_in[4];
    const int* mapped_idx = (const int*)d_in[5];

    const int N  = in_sizes[0];
    const int B1 = in_sizes[3];   // B + 1
    const int V  = in_sizes[4];
    const int S  = (V + PIVOTS - 1) / PIVOTS;

    // Output flat layout (return order): offsets | idx_values | offsets | idx_keys | weights
    int* out           = (int*)d_out;
    int* out_offsets0  = out;
    int* out_values    = out + B1;
    int* out_offsets1  = out + B1 + N;
    int* out_keys      = out + 2 * B1 + N;
    float* out_weights = (float*)(out + 2 * B1 + 2 * N);

    // Pass-through copies at full HBM bandwidth (graph-capture safe)
    hipMemcpyAsync(out_offsets0, offsets, (size_t)B1 * sizeof(int),
                   hipMemcpyDeviceToDevice, stream);
    hipMemcpyAsync(out_offsets1, offsets, (size_t)B1 * sizeof(int),
                   hipMemcpyDeviceToDevice, stream);
    hipMemcpyAsync(out_weights, scores, (size_t)N * sizeof(float),
                   hipMemcpyDeviceToDevice, stream);

    sparse_remap2_kernel<<<GRID, BLOCK, 0, stream>>>(
        raw_values, raw_keys, sorted_ids, mapped_idx,
        out_values, out_keys, N, V, S);
}